// PerturbedLoss_80590766342735
// MI455X (gfx1250) — compile-verified
//
#include <hip/hip_runtime.h>

// Problem dims (fixed by the reference).
#define S_SAMPLES 64
#define BATCH     128
#define DIM       10000
#define NV4       2500            // DIM / 4 float4 chunks per row
#define TPB       256             // 8 wave32s
#define KMAX      10              // ceil(NV4 / TPB)
#define NROWS     (S_SAMPLES * BATCH)

typedef float v4f __attribute__((ext_vector_type(4)));

// One block per (s,b) row: softmax over D + MSE vs y, partial sum -> part[row].
__global__ __launch_bounds__(TPB) void ptb_row_kernel(
    const float* __restrict__ theta,
    const float* __restrict__ y,
    const float* __restrict__ noise,
    float* __restrict__ part)
{
    __shared__ v4f   lds_y[NV4];          // 40 KB staged y row
    __shared__ float red[TPB / 32];
    __shared__ float s_inv;

    const int tid = threadIdx.x;
    const int row = blockIdx.x;           // row = s*BATCH + b
    const int b   = row % BATCH;

    const v4f* __restrict__ nrow = (const v4f*)(noise + (size_t)row * DIM);
    const v4f* __restrict__ trow = (const v4f*)(theta + (size_t)b   * DIM);
    const unsigned long long ybase =
        (unsigned long long)(const void*)(y + (size_t)b * DIM);

    // Generic LDS pointer low 32 bits == wave-relative LDS byte offset (gfx1250).
    const unsigned lds_base = (unsigned)(size_t)(&lds_y[0]);

    // ---- Issue async y-row -> LDS (overlaps with exp pass; ASYNCcnt tracked) ----
    #pragma unroll
    for (int k = 0; k < KMAX; ++k) {
        int i = tid + k * TPB;
        if (i < NV4) {
            unsigned laddr = lds_base + (unsigned)i * 16u;  // LDS byte addr (16B aligned)
            unsigned goff  = (unsigned)i * 16u;             // byte offset into row
            // GVS mode: mem = SGPR64 base + VGPR32 offset
            asm volatile("global_load_async_to_lds_b128 %0, %1, %2"
                         :: "v"(laddr), "v"(goff), "s"(ybase)
                         : "memory");
        }
    }

    // ---- Pass A: e = exp(theta + noise), keep in registers, accumulate sum ----
    v4f   e[KMAX];
    float lsum = 0.0f;
    #pragma unroll
    for (int k = 0; k < KMAX; ++k) {
        int i = tid + k * TPB;
        if (i < NV4) {
            v4f nz = __builtin_nontemporal_load(nrow + i);  // streamed once: TH=NT
            v4f th = trow[i];                               // L2-resident, cached
            v4f ev;
            ev.x = __expf(th.x + nz.x);
            ev.y = __expf(th.y + nz.y);
            ev.z = __expf(th.z + nz.z);
            ev.w = __expf(th.w + nz.w);
            e[k] = ev;
            lsum += (ev.x + ev.y) + (ev.z + ev.w);
        }
    }

    // Block reduction of the softmax denominator (wave32 butterfly + LDS).
    #pragma unroll
    for (int off = 16; off > 0; off >>= 1) lsum += __shfl_xor(lsum, off, 32);
    const int wave = tid >> 5, lane = tid & 31;
    if (lane == 0) red[wave] = lsum;
    __syncthreads();
    if (tid == 0) {
        float t = 0.0f;
        #pragma unroll
        for (int w = 0; w < TPB / 32; ++w) t += red[w];
        s_inv = 1.0f / t;
    }
    __syncthreads();
    const float inv = s_inv;

    // ---- Wait for async y data (this wave issued its own lanes' loads) ----
    asm volatile("s_wait_asynccnt 0x0" ::: "memory");

    // ---- Pass B: MSE accumulation, pi = e * inv ----
    float lacc = 0.0f;
    #pragma unroll
    for (int k = 0; k < KMAX; ++k) {
        int i = tid + k * TPB;
        if (i < NV4) {
            v4f yv = lds_y[i];
            v4f d;
            d.x = e[k].x * inv - yv.x;
            d.y = e[k].y * inv - yv.y;
            d.z = e[k].z * inv - yv.z;
            d.w = e[k].w * inv - yv.w;
            lacc += (d.x * d.x + d.y * d.y) + (d.z * d.z + d.w * d.w);
        }
    }

    #pragma unroll
    for (int off = 16; off > 0; off >>= 1) lacc += __shfl_xor(lacc, off, 32);
    if (lane == 0) red[wave] = lacc;
    __syncthreads();
    if (tid == 0) {
        float t = 0.0f;
        #pragma unroll
        for (int w = 0; w < TPB / 32; ++w) t += red[w];
        part[row] = t;   // sum of squared errors for this row (un-normalized)
    }
}

// Deterministic final reduction: 8192 partials -> scalar mean.
__global__ __launch_bounds__(TPB) void ptb_final_kernel(
    const float* __restrict__ part, float* __restrict__ out)
{
    __shared__ float red[TPB / 32];
    const int tid = threadIdx.x;
    float s = 0.0f;
    for (int i = tid; i < NROWS; i += TPB) s += part[i];
    #pragma unroll
    for (int off = 16; off > 0; off >>= 1) s += __shfl_xor(s, off, 32);
    if ((tid & 31) == 0) red[tid >> 5] = s;
    __syncthreads();
    if (tid == 0) {
        float t = 0.0f;
        #pragma unroll
        for (int w = 0; w < TPB / 32; ++w) t += red[w];
        out[0] = t * (1.0f / (float)((long long)S_SAMPLES * BATCH * DIM));
    }
}

extern "C" void kernel_launch(void* const* d_in, const int* in_sizes, int n_in,
                              void* d_out, int out_size, void* d_ws, size_t ws_size,
                              hipStream_t stream) {
    const float* theta = (const float*)d_in[0];
    const float* y     = (const float*)d_in[1];
    const float* noise = (const float*)d_in[2];
    float* part = (float*)d_ws;   // NROWS floats = 32 KB scratch

    ptb_row_kernel<<<NROWS, TPB, 0, stream>>>(theta, y, noise, part);
    ptb_final_kernel<<<1, TPB, 0, stream>>>(part, (float*)d_out);
}